// Attention_34780645163058
// MI455X (gfx1250) — compile-verified
//
#include <hip/hip_runtime.h>
#include <math.h>

// Problem constants (from reference): B=1024, N=512, E=256, H=128, A=32
constexpr int kB = 1024;
constexpr int kN = 512;
constexpr int kE = 256;
constexpr int kH = 128;
constexpr int kA = 32;

typedef __attribute__((ext_vector_type(2))) float v2f;
typedef __attribute__((ext_vector_type(8))) float v8f;
typedef __attribute__((ext_vector_type(4))) unsigned int u32x4;
typedef __attribute__((ext_vector_type(4))) int i32x4;
typedef __attribute__((ext_vector_type(8))) int i32x8;

// ---------------------------------------------------------------------------
// Kernel 1: dproj[b,h] = b1[h] + b2[h] + sum_k dec[b,k] * W2[k,h]
// ---------------------------------------------------------------------------
__global__ __launch_bounds__(kH)
void k_decproj(const float* __restrict__ dec, const float* __restrict__ W2,
               const float* __restrict__ b1, const float* __restrict__ b2,
               float* __restrict__ dproj) {
  __shared__ float sdec[kH];
  const int b = blockIdx.x;
  const int h = threadIdx.x;
  sdec[h] = dec[b * kH + h];
  __syncthreads();
  float s = b1[h] + b2[h];
#pragma unroll 8
  for (int k = 0; k < kH; ++k) s = fmaf(sdec[k], W2[k * kH + h], s);
  dproj[b * kH + h] = s;
}

// ---------------------------------------------------------------------------
// Kernel 2: x_t = enc^T·W1 + dproj ; u_t = tanh(x_t·Wa + ba)·Va + bva
// Block = 256 threads (8 waves), one (b, 32-row n-tile) per block.
// TDM stages the enc tile [256e x 32n] into LDS (pitch 33 via TDM padding);
// wave w: 16-row half (w&1), 32-col H slice ((w>>1)*32), 2 wmma per K-step.
// ---------------------------------------------------------------------------
__global__ __launch_bounds__(256)
void k_xt_ut(const float* __restrict__ enc, const float* __restrict__ W1,
             const float* __restrict__ dproj, const float* __restrict__ Wa,
             const float* __restrict__ ba, const float* __restrict__ Va,
             const float* __restrict__ bva,
             float* __restrict__ xt, float* __restrict__ ut) {
  constexpr int TN = 32;        // n rows per block
  constexpr int LP = TN + 1;    // LDS pitch (TDM pads 1 dword per 32 dwords)
  const int tile = blockIdx.x;
  const int b    = tile >> 4;           // N/32 = 16 tiles per batch
  const int n0   = (tile & 15) << 5;
  const int lane = threadIdx.x & 31;
  const int w    = threadIdx.x >> 5;
  const int nsub = w & 1;
  const int h0   = (w >> 1) << 5;
  const int m    = lane & 15;
  const int krow = (lane >> 4) << 1;    // K sub-offset per ISA f32 A/B layout

  __shared__ float sE[kE * LP];         // enc tile, [e][n] row-major + pad
  __shared__ float sx[TN][kH + 4];
  __shared__ float red2[TN][8];

  // ---- TDM: DMA enc[b, 0:256, n0:n0+32] -> LDS (D# per ISA 8.3/8.4) ----
  if (w == 0) {
    const unsigned long long ga =
        (unsigned long long)(const void*)(enc + (size_t)b * kE * kN + n0);
    const unsigned lds = (unsigned)(size_t)(void*)&sE[0];
    u32x4 g0;
    g0.x = 1u;                                   // count = 1 valid descriptor
    g0.y = lds;                                  // lds_addr (bytes)
    g0.z = (unsigned)(ga & 0xffffffffu);         // global_addr[31:0]
    g0.w = (unsigned)((ga >> 32) & 0x01ffffffu)  // global_addr[56:32]
           | (2u << 30);                         // type = 2 ("image")
    i32x8 g1;
    g1[0] = (int)((2u << 16)      // data_size = 4 bytes
                | (1u << 20)      // pad_enable
                | (4u << 22));    // pad_interval: every 32 dwords (+1 dword)
    g1[1] = (int)(((unsigned)kN & 0xffffu) << 16);  // tensor_dim0 = 512 (lo16)
    g1[2] = (int)(((unsigned)kE & 0xffffu) << 16);  // dim0 hi | tensor_dim1 = 256
    g1[3] = (int)((unsigned)TN << 16);              // dim1 hi | tile_dim0 = 32
    g1[4] = (int)(unsigned)kE;                      // tile_dim1 = 256 | tile_dim2 = 0
    g1[5] = (int)(unsigned)kN;                      // tensor_dim0_stride = 512
    g1[6] = 0;                                      // stride hi | dim1_stride lo
    g1[7] = 0;
    i32x4 z4 = {0, 0, 0, 0};
#if defined(__clang_major__) && __clang_major__ >= 23
    i32x8 z8 = {0, 0, 0, 0, 0, 0, 0, 0};
    __builtin_amdgcn_tensor_load_to_lds(g0, g1, z4, z4, z8, 0);
#else
    __builtin_amdgcn_tensor_load_to_lds(g0, g1, z4, z4, 0);
#endif
    __builtin_amdgcn_s_wait_tensorcnt(0);
  }
  __syncthreads();

  // ---- WMMA: A from LDS (1 fragment -> 2 wmma), B = W1 from L2 ----
  const int colA = nsub * 16 + m;
  const float* bp0 = W1 + h0 + m;
  const float* bp1 = W1 + h0 + 16 + m;
  v8f acc0 = {}, acc1 = {};
#pragma unroll 4
  for (int k0 = 0; k0 < kE; k0 += 4) {
    const int kk = k0 + krow;
    v2f av, bv0, bv1;
    av.x  = sE[kk * LP + colA];
    av.y  = sE[(kk + 1) * LP + colA];
    bv0.x = bp0[kk * kH];
    bv0.y = bp0[(kk + 1) * kH];
    bv1.x = bp1[kk * kH];
    bv1.y = bp1[(kk + 1) * kH];
    acc0 = __builtin_amdgcn_wmma_f32_16x16x4_f32(false, av, false, bv0,
                                                 (short)0, acc0, false, false);
    acc1 = __builtin_amdgcn_wmma_f32_16x16x4_f32(false, av, false, bv1,
                                                 (short)0, acc1, false, false);
  }

  const int   rbase = (lane >> 4) << 3;
  const float dp0 = dproj[b * kH + h0 + m];
  const float dp1 = dproj[b * kH + h0 + 16 + m];
#pragma unroll
  for (int r = 0; r < 8; ++r) {
    const int row = nsub * 16 + rbase + r;
    const float v0 = acc0[r] + dp0;
    const float v1 = acc1[r] + dp1;
    sx[row][h0 + m]      = v0;
    sx[row][h0 + 16 + m] = v1;
    float* xrow = xt + ((size_t)b * kN + n0 + row) * kH;
    xrow[h0 + m]      = v0;
    xrow[h0 + 16 + m] = v1;
  }
  __syncthreads();

  // ---- u_t: thread t -> row t&31, 4 A-columns starting 4*(t>>5) ----
  const int row = threadIdx.x & 31;
  const int ac  = (threadIdx.x >> 5) << 2;
  float s0 = ba[ac], s1 = ba[ac + 1], s2 = ba[ac + 2], s3 = ba[ac + 3];
#pragma unroll 4
  for (int h = 0; h < kH; ++h) {
    const float xv = sx[row][h];
    s0 = fmaf(xv, Wa[h * kA + ac],     s0);
    s1 = fmaf(xv, Wa[h * kA + ac + 1], s1);
    s2 = fmaf(xv, Wa[h * kA + ac + 2], s2);
    s3 = fmaf(xv, Wa[h * kA + ac + 3], s3);
  }
  red2[row][threadIdx.x >> 5] =
      tanhf(s0) * Va[ac] + tanhf(s1) * Va[ac + 1] +
      tanhf(s2) * Va[ac + 2] + tanhf(s3) * Va[ac + 3];
  __syncthreads();
  if (threadIdx.x < TN) {
    float u = bva[0];
#pragma unroll
    for (int i = 0; i < 8; ++i) u += red2[threadIdx.x][i];
    ut[(size_t)b * kN + n0 + threadIdx.x] = u;
  }
}

// ---------------------------------------------------------------------------
// Kernel 3: per-b softmax(u_t) -> a_t (LDS only), c_t = sum_n a_t*x_t,
//           cw[b,h] = bct[h] + bwt[h] + c_t·Wct[:,h]
// ---------------------------------------------------------------------------
__global__ __launch_bounds__(512)
void k_ctx(const float* __restrict__ ut, const float* __restrict__ xt,
           const float* __restrict__ Wct, const float* __restrict__ bct,
           const float* __restrict__ bwt, float* __restrict__ cw) {
  const int b = blockIdx.x;
  const int t = threadIdx.x;
  __shared__ float sa[kN];
  __shared__ float sred[512];
  __shared__ float sc[kH];

  const float u = ut[(size_t)b * kN + t];
  sred[t] = u;
  __syncthreads();
  for (int off = 256; off > 0; off >>= 1) {
    if (t < off) sred[t] = fmaxf(sred[t], sred[t + off]);
    __syncthreads();
  }
  const float mx = sred[0];
  __syncthreads();
  const float e = expf(u - mx);
  sred[t] = e;
  __syncthreads();
  for (int off = 256; off > 0; off >>= 1) {
    if (t < off) sred[t] += sred[t + off];
    __syncthreads();
  }
  const float inv = 1.0f / sred[0];
  __syncthreads();
  sa[t] = e * inv;
  __syncthreads();

  const int h = t & (kH - 1);
  const int q = t >> 7;
  float acc = 0.0f;
  const float* xb = xt + ((size_t)b * kN + q * 128) * kH + h;
#pragma unroll 4
  for (int n = 0; n < 128; ++n) acc = fmaf(sa[q * 128 + n], xb[(size_t)n * kH], acc);
  sred[t] = acc;
  __syncthreads();
  if (t < kH) sc[t] = sred[t] + sred[t + 128] + sred[t + 256] + sred[t + 384];
  __syncthreads();
  if (t < kH) {
    float s = bct[t] + bwt[t];
#pragma unroll 8
    for (int k = 0; k < kH; ++k) s = fmaf(sc[k], Wct[k * kH + t], s);
    cw[b * kH + t] = s;
  }
}

// ---------------------------------------------------------------------------
// Kernel 4: w_c = x_t·Wwt + cw[b,:] (WMMA, K=H=128, 2 H-tiles per wave)
//           lg[b,n] = tanh(w_c)·Vc + bvc  (deterministic shuffle+LDS reduce)
// Block = 256 threads, one (b, 32-row n-tile) per block.
// ---------------------------------------------------------------------------
__global__ __launch_bounds__(256)
void k_wc_logits(const float* __restrict__ xt, const float* __restrict__ Wwt,
                 const float* __restrict__ cw, const float* __restrict__ Vc,
                 const float* __restrict__ bvc, float* __restrict__ lg) {
  const int tile = blockIdx.x;
  const int b    = tile >> 4;
  const int n0   = (tile & 15) << 5;
  const int lane = threadIdx.x & 31;
  const int w    = threadIdx.x >> 5;
  const int nsub = w & 1;
  const int h0   = (w >> 1) << 5;
  const int m    = lane & 15;
  const int krow = (lane >> 4) << 1;

  __shared__ float red[4][32];

  const float* aRow = xt + ((size_t)b * kN + n0 + nsub * 16 + m) * kH;
  const float* bp0  = Wwt + h0 + m;
  const float* bp1  = Wwt + h0 + 16 + m;

  v8f acc0 = {}, acc1 = {};
#pragma unroll 4
  for (int k0 = 0; k0 < kH; k0 += 4) {
    const int kk = k0 + krow;
    const v2f av = *(const v2f*)(aRow + kk);      // contiguous K, 8B aligned
    v2f bv0, bv1;
    bv0.x = bp0[kk * kH];
    bv0.y = bp0[(kk + 1) * kH];
    bv1.x = bp1[kk * kH];
    bv1.y = bp1[(kk + 1) * kH];
    acc0 = __builtin_amdgcn_wmma_f32_16x16x4_f32(false, av, false, bv0,
                                                 (short)0, acc0, false, false);
    acc1 = __builtin_amdgcn_wmma_f32_16x16x4_f32(false, av, false, bv1,
                                                 (short)0, acc1, false, false);
  }

  const int   rbase = (lane >> 4) << 3;
  const float cw0 = cw[b * kH + h0 + m];
  const float cw1 = cw[b * kH + h0 + 16 + m];
  const float vc0 = Vc[h0 + m];
  const float vc1 = Vc[h0 + 16 + m];
#pragma unroll
  for (int r = 0; r < 8; ++r) {
    float val = tanhf(acc0[r] + cw0) * vc0 + tanhf(acc1[r] + cw1) * vc1;
    val += __shfl_xor(val, 1, 16);
    val += __shfl_xor(val, 2, 16);
    val += __shfl_xor(val, 4, 16);
    val += __shfl_xor(val, 8, 16);
    if (m == 0) red[w >> 1][nsub * 16 + rbase + r] = val;
  }
  __syncthreads();
  if (threadIdx.x < 32) {
    float s = bvc[0];
#pragma unroll
    for (int i = 0; i < 4; ++i) s += red[i][threadIdx.x];
    lg[(size_t)b * kN + n0 + threadIdx.x] = s;
  }
}

// ---------------------------------------------------------------------------
// Kernel 5: log_softmax, mask, argmax (first-index ties), demand scatter.
// ---------------------------------------------------------------------------
__global__ __launch_bounds__(512)
void k_final(const float* __restrict__ lg, const float* __restrict__ demand,
             const float* __restrict__ load, const int* __restrict__ vcap,
             float* __restrict__ out_lp, float* __restrict__ out_ptr,
             float* __restrict__ out_load, float* __restrict__ out_dem) {
  const int b = blockIdx.x;
  const int n = threadIdx.x;
  __shared__ float sred[512];
  __shared__ int   sidx[512];
  __shared__ int   sp;
  __shared__ float snd;

  const float l = lg[(size_t)b * kN + n];
  sred[n] = l;
  __syncthreads();
  for (int off = 256; off > 0; off >>= 1) {
    if (n < off) sred[n] = fmaxf(sred[n], sred[n + off]);
    __syncthreads();
  }
  const float mx = sred[0];
  __syncthreads();
  sred[n] = expf(l - mx);
  __syncthreads();
  for (int off = 256; off > 0; off >>= 1) {
    if (n < off) sred[n] += sred[n + off];
    __syncthreads();
  }
  const float lse = mx + logf(sred[0]);
  __syncthreads();

  float lp = l - lse;
  const float ld = load[b];
  const float dm = demand[(size_t)b * kN + n];
  if (n >= 1 && (dm == 0.0f || ld == 0.0f)) lp = -INFINITY;
  out_lp[(size_t)b * kN + n] = lp;

  sred[n] = lp;
  sidx[n] = n;
  __syncthreads();
  for (int off = 256; off > 0; off >>= 1) {
    if (n < off) {
      const float v2 = sred[n + off];
      const int   i2 = sidx[n + off];
      if (v2 > sred[n] || (v2 == sred[n] && i2 < sidx[n])) {
        sred[n] = v2;
        sidx[n] = i2;
      }
    }
    __syncthreads();
  }
  if (n == 0) {
    const int   p     = sidx[0];
    const float dsel  = demand[(size_t)b * kN + p];
    const float cap   = (float)(*vcap);
    const bool  depot = (p == 0);
    const float nl    = depot ? cap : (ld - dsel);
    const float nds   = depot ? dsel : (dsel - nl);
    out_ptr[b]  = (float)p;
    out_load[b] = nl;
    sp  = p;
    snd = nds;
  }
  __syncthreads();
  out_dem[(size_t)b * kN + n] = (n == sp) ? snd : dm;
}

// ---------------------------------------------------------------------------
extern "C" void kernel_launch(void* const* d_in, const int* in_sizes, int n_in,
                              void* d_out, int out_size, void* d_ws, size_t ws_size,
                              hipStream_t stream) {
  const float* dec  = (const float*)d_in[0];
  const float* enc  = (const float*)d_in[1];
  const float* load = (const float*)d_in[2];
  const float* dem  = (const float*)d_in[3];
  const float* W1   = (const float*)d_in[4];
  const float* b1   = (const float*)d_in[5];
  const float* W2   = (const float*)d_in[6];
  const float* b2   = (const float*)d_in[7];
  const float* Wwt  = (const float*)d_in[8];
  const float* bwt  = (const float*)d_in[9];
  const float* Wct  = (const float*)d_in[10];
  const float* bct  = (const float*)d_in[11];
  const float* Wa   = (const float*)d_in[12];
  const float* ba   = (const float*)d_in[13];
  const float* Va   = (const float*)d_in[14];
  const float* bva  = (const float*)d_in[15];
  const float* Vc   = (const float*)d_in[16];
  const float* bvc  = (const float*)d_in[17];
  const int*   vcap = (const int*)d_in[18];

  float* ws    = (float*)d_ws;
  float* dproj = ws;                               // B*H
  float* xt    = dproj + (size_t)kB * kH;          // B*N*H (256 MB)
  float* ut    = xt + (size_t)kB * kN * kH;        // B*N
  float* cw    = ut + (size_t)kB * kN;             // B*H
  float* lg    = cw + (size_t)kB * kH;             // B*N

  float* out      = (float*)d_out;
  float* out_lp   = out;                           // B*N
  float* out_ptr  = out_lp + (size_t)kB * kN;      // B
  float* out_load = out_ptr + kB;                  // B
  float* out_dem  = out_load + kB;                 // B*N

  k_decproj<<<kB, kH, 0, stream>>>(dec, W2, b1, b2, dproj);
  k_xt_ut<<<kB * (kN / 32), 256, 0, stream>>>(enc, W1, dproj, Wa, ba, Va, bva, xt, ut);
  k_ctx<<<kB, 512, 0, stream>>>(ut, xt, Wct, bct, bwt, cw);
  k_wc_logits<<<kB * (kN / 32), 256, 0, stream>>>(xt, Wwt, cw, Vc, bvc, lg);
  k_final<<<kB, 512, 0, stream>>>(lg, dem, load, vcap, out_lp, out_ptr, out_load, out_dem);
}